// MultiBoxLoss_678604833038
// MI455X (gfx1250) — compile-verified
//
#include <hip/hip_runtime.h>
#include <hip/hip_bf16.h>

#define NB 16
#define NA 65472
#define NC 81
#define NG 50
#define POS_THRESH 0.5f
#define BETA (1.0f / 9.0f)
#define POSKEY 0x007FFFFFu   // key_desc(-inf)

typedef __attribute__((ext_vector_type(2))) float v2f;
typedef __attribute__((ext_vector_type(8))) float v8f;

// ---------------- wave32 f32 sum via V_WMMA_F32_16X16X4_F32 ----------------
// A (16x4): VGPR0 holds K=0 (lanes 0-15) / K=2 (lanes 16-31); VGPR1 -> 0.
// B (4x16): all ones.  D[m,n] = v_m + v_{m+16}.  Per-lane sum of the 8 D regs
// gives half-sums; shfl_xor(16) completes the full wave sum (broadcast).
__device__ __forceinline__ float wave_sum_wmma(float v) {
  v2f a; a[0] = v;    a[1] = 0.0f;
  v2f o; o[0] = 1.0f; o[1] = 1.0f;
  v8f c = {};
  c = __builtin_amdgcn_wmma_f32_16x16x4_f32(false, a, false, o, (short)0, c,
                                            false, false);
  float s = c[0] + c[1] + c[2] + c[3] + c[4] + c[5] + c[6] + c[7];
  s += __shfl_xor(s, 16, 32);
  return s;
}

__device__ __forceinline__ float iou_box(float4 p, float4 q) {
  float lx = fmaxf(p.x, q.x), ly = fmaxf(p.y, q.y);
  float rx = fminf(p.z, q.z), ry = fminf(p.w, q.w);
  float w = fmaxf(rx - lx, 0.0f), h = fmaxf(ry - ly, 0.0f);
  float inter = w * h;
  float areaA = (p.z - p.x) * (p.w - p.y);
  float areaB = (q.z - q.x) * (q.w - q.y);
  return inter / (areaA + areaB - inter + 1e-10f);
}

__device__ __forceinline__ unsigned key_desc(float f) {
  unsigned u = __float_as_uint(f);
  return (u & 0x80000000u) ? ~u : (u | 0x80000000u);
}

// ---------------- K0: zero accumulators ----------------
__global__ void k_init(unsigned* posCount, unsigned* totalPos, float* locSum,
                       float* classSum) {
  int t = blockIdx.x * blockDim.x + threadIdx.x;
  if (t < NB) posCount[t] = 0u;
  if (t == 0) { *totalPos = 0u; *locSum = 0.0f; *classSum = 0.0f; }
}

// ---------------- K1: per-anchor best IoU / argmax over gts ----------------
__global__ void k_match_anchor(const float* __restrict__ gts,
                               const int* __restrict__ counts,
                               const float* __restrict__ anchors,
                               float* __restrict__ bestIou,
                               int* __restrict__ bestIdx) {
  int b = blockIdx.y;
  int a = blockIdx.x * blockDim.x + threadIdx.x;
  __shared__ float4 sbox[NG];
  int cnt = counts[b];
  if (threadIdx.x < NG) {
    const float* g = gts + ((size_t)b * NG + threadIdx.x) * 5;
    sbox[threadIdx.x] = make_float4(g[0], g[1], g[2], g[3]);
  }
  __syncthreads();
  if (a >= NA) return;
  float4 p = *(const float4*)(anchors + (size_t)a * 4);
  float best = -__builtin_inff();
  int bidx = 0;
  for (int g = 0; g < NG; ++g) {
    float v = (g < cnt) ? iou_box(p, sbox[g]) : -1.0f;
    if (v > best) { best = v; bidx = g; }
  }
  size_t i = (size_t)b * NA + a;
  bestIou[i] = best;
  bestIdx[i] = bidx;
}

// ---------------- K1b: per-gt best anchor (argmax over A) ----------------
__global__ void k_best_anchor(const float* __restrict__ gts,
                              const int* __restrict__ counts,
                              const float* __restrict__ anchors,
                              int* __restrict__ bestAnchor) {
  int b = blockIdx.y, g = blockIdx.x;
  __shared__ unsigned long long sk[256];
  if (g >= counts[b]) {
    if (threadIdx.x == 0) bestAnchor[b * NG + g] = -1;
    return;
  }
  const float* gb = gts + ((size_t)b * NG + g) * 5;
  float4 q = make_float4(gb[0], gb[1], gb[2], gb[3]);
  unsigned long long best = 0ull;  // valid IoU >= 0, key strictly positive
  for (int a = threadIdx.x; a < NA; a += blockDim.x) {
    float4 p = *(const float4*)(anchors + (size_t)a * 4);
    float v = iou_box(p, q);
    // tie-break: argmax keeps lowest anchor index -> pack ~a in LSBs
    unsigned long long key = ((unsigned long long)__float_as_uint(v) << 32) |
                             (unsigned)(0xFFFFFFFFu - (unsigned)a);
    if (key > best) best = key;
  }
  sk[threadIdx.x] = best;
  __syncthreads();
  for (int s = 128; s > 0; s >>= 1) {
    if (threadIdx.x < s)
      sk[threadIdx.x] =
          sk[threadIdx.x] > sk[threadIdx.x + s] ? sk[threadIdx.x] : sk[threadIdx.x + s];
    __syncthreads();
  }
  if (threadIdx.x == 0)
    bestAnchor[b * NG + g] = (int)(0xFFFFFFFFu - (unsigned)(sk[0] & 0xFFFFFFFFull));
}

// ---------------- K2: scatter best-anchor overrides ----------------
__global__ void k_scatter(const int* __restrict__ counts,
                          const int* __restrict__ bestAnchor,
                          float* __restrict__ bestIou, int* __restrict__ bestIdx) {
  int b = blockIdx.x;
  if (threadIdx.x != 0) return;
  int cnt = counts[b];
  for (int g = 0; g < cnt; ++g) {   // serial: last write wins (matches scatter)
    int a = bestAnchor[b * NG + g];
    if (a >= 0 && a < NA) {
      size_t i = (size_t)b * NA + a;
      bestIdx[i] = g;
      bestIou[i] = 2.0f;
    }
  }
}

// ---------------- K3: smooth-L1 loc loss + positive counts ----------------
__global__ void k_locpos(const float* __restrict__ gts,
                         const float* __restrict__ anchors,
                         const float* __restrict__ ploc,
                         const float* __restrict__ bestIou,
                         const int* __restrict__ bestIdx,
                         unsigned* __restrict__ posCount,
                         unsigned* __restrict__ totalPos,
                         float* __restrict__ locSum) {
  int b = blockIdx.y;
  int a = blockIdx.x * blockDim.x + threadIdx.x;
  float contrib = 0.0f;
  bool pos = false;
  if (a < NA) {
    size_t i = (size_t)b * NA + a;
    float iou = bestIou[i];
    int idx = bestIdx[i];
    const float* g = gts + ((size_t)b * NG + idx) * 5;
    int label = (iou < POS_THRESH) ? 0 : (int)g[4];
    pos = label > 0;
    if (pos) {
      float ax0 = anchors[a * 4 + 0], ay0 = anchors[a * 4 + 1];
      float ax1 = anchors[a * 4 + 2], ay1 = anchors[a * 4 + 3];
      float acx = (ax0 + ax1) * 0.5f, acy = (ay0 + ay1) * 0.5f;
      float aw = ax1 - ax0, ah = ay1 - ay0;
      float mcx = (g[0] + g[2]) * 0.5f, mcy = (g[1] + g[3]) * 0.5f;
      float mw = g[2] - g[0], mh = g[3] - g[1];
      float t[4];
      t[0] = (mcx - acx) / (aw * 0.1f);
      t[1] = (mcy - acy) / (ah * 0.1f);
      t[2] = logf(mw / aw + 1e-10f) / 0.2f;
      t[3] = logf(mh / ah + 1e-10f) / 0.2f;
      const float* p = ploc + i * 4;
      for (int j = 0; j < 4; ++j) {
        float n = fabsf(p[j] - t[j]);
        contrib += (n < BETA) ? 0.5f * n * n / BETA : n - 0.5f * BETA;
      }
    }
  }
  unsigned long long bal = __ballot(pos);
  float s = wave_sum_wmma(contrib);     // all 32 lanes converged here
  if ((threadIdx.x & 31) == 0) {
    unsigned c = (unsigned)__popcll(bal);
    if (c) { atomicAdd(&posCount[b], c); atomicAdd(totalPos, c); }
    if (s != 0.0f) atomicAdd(locSum, s);
  }
}

// ---------------- K4: wave-per-anchor logsumexp, CE, mining key ----------------
__global__ void k_lse(const float* __restrict__ conf, const float* __restrict__ gts,
                      const float* __restrict__ bestIou,
                      const int* __restrict__ bestIdx, float* __restrict__ ce,
                      unsigned* __restrict__ mkey) {
  int b = blockIdx.y;
  int wave = threadIdx.x >> 5;
  int lane = threadIdx.x & 31;
  int a = blockIdx.x * 8 + wave;
  if (a >= NA) return;
  size_t row = ((size_t)b * NA + a) * NC;
  __builtin_prefetch(conf + row + NC, 0, 0);  // global_prefetch_b8: next row
  float x0 = conf[row + lane];                               // lane < 81 always
  float x1 = (lane + 32 < NC) ? conf[row + lane + 32] : -__builtin_inff();
  float x2 = (lane + 64 < NC) ? conf[row + lane + 64] : -__builtin_inff();
  float m = fmaxf(x0, fmaxf(x1, x2));
  for (int o = 16; o > 0; o >>= 1) m = fmaxf(m, __shfl_xor(m, o, 32));
  float s = expf(x0 - m);
  if (lane + 32 < NC) s += expf(x1 - m);
  if (lane + 64 < NC) s += expf(x2 - m);
  for (int o = 16; o > 0; o >>= 1) s += __shfl_xor(s, o, 32);
  if (lane == 0) {
    float lse = m + logf(s);
    size_t i = (size_t)b * NA + a;
    float iou = bestIou[i];
    int idx = bestIdx[i];
    int label = (iou < POS_THRESH) ? 0 : (int)gts[((size_t)b * NG + idx) * 5 + 4];
    ce[i] = lse - conf[row + label];
    float mining = (label > 0) ? -__builtin_inff() : (lse - conf[row]);
    mkey[i] = key_desc(mining);
  }
}

// ---------------- K5: per-batch radix select of (3*num_pos)-th largest key ----
__global__ void k_select(const unsigned* __restrict__ mkey,
                         const unsigned* __restrict__ posCount,
                         unsigned* __restrict__ kthKey, unsigned* __restrict__ kPos) {
  int b = blockIdx.x;
  __shared__ unsigned hist[256];
  __shared__ unsigned s_prefix, s_mask, s_k;
  unsigned k = 3u * posCount[b];
  if (threadIdx.x == 0) { kPos[b] = k; s_prefix = 0u; s_mask = 0u; s_k = k; }
  __syncthreads();
  if (k == 0u) { if (threadIdx.x == 0) kthKey[b] = 0xFFFFFFFFu; return; }
  if (k >= (unsigned)NA) { if (threadIdx.x == 0) kthKey[b] = 0u; return; }
  const unsigned* keys = mkey + (size_t)b * NA;
  for (int p = 3; p >= 0; --p) {
    hist[threadIdx.x] = 0u;
    __syncthreads();
    unsigned prefix = s_prefix, mask = s_mask;
    for (int i = threadIdx.x; i < NA; i += 256) {
      unsigned key = keys[i];
      if ((key & mask) == prefix)
        atomicAdd(&hist[(key >> (8 * p)) & 0xFFu], 1u);
    }
    __syncthreads();
    if (threadIdx.x == 0) {
      unsigned kk = s_k;
      int bin = 255;
      for (; bin > 0; --bin) {
        unsigned h = hist[bin];
        if (kk <= h) break;
        kk -= h;
      }
      s_k = kk;
      s_prefix = prefix | ((unsigned)bin << (8 * p));
      s_mask = mask | (0xFFu << (8 * p));
    }
    __syncthreads();
  }
  if (threadIdx.x == 0) kthKey[b] = s_prefix;
}

// ---------------- K6: masked CE sum ----------------
__global__ void k_classsum(const float* __restrict__ ce,
                           const unsigned* __restrict__ mkey,
                           const unsigned* __restrict__ kthKey,
                           const unsigned* __restrict__ kPos,
                           float* __restrict__ classSum) {
  int b = blockIdx.y;
  int a = blockIdx.x * blockDim.x + threadIdx.x;
  float v = 0.0f;
  if (a < NA) {
    size_t i = (size_t)b * NA + a;
    unsigned key = mkey[i];
    bool pos = (key == POSKEY);
    bool neg = (kPos[b] > 0u) && (key >= kthKey[b]) && !pos;
    if (pos || neg) v = ce[i];
  }
  float s = wave_sum_wmma(v);
  if ((threadIdx.x & 31) == 0 && s != 0.0f) atomicAdd(classSum, s);
}

// ---------------- K7: finalize ----------------
__global__ void k_final(const unsigned* __restrict__ totalPos,
                        const float* __restrict__ locSum,
                        const float* __restrict__ classSum, float* __restrict__ out) {
  if (blockIdx.x == 0 && threadIdx.x == 0) {
    float np = fmaxf(1.0f, (float)(*totalPos));
    out[0] = *locSum / (np * 4.0f);    // localisation_loss
    out[1] = *classSum / (np * 4.0f);  // classification_loss
  }
}

extern "C" void kernel_launch(void* const* d_in, const int* in_sizes, int n_in,
                              void* d_out, int out_size, void* d_ws, size_t ws_size,
                              hipStream_t stream) {
  const float* conf    = (const float*)d_in[0];
  const float* ploc    = (const float*)d_in[1];
  const float* gts     = (const float*)d_in[2];
  const int*   counts  = (const int*)d_in[3];
  const float* anchors = (const float*)d_in[4];
  float* out = (float*)d_out;

  const size_t BA = (size_t)NB * NA;
  char* ws = (char*)d_ws;
  size_t off = 0;
  auto take = [&](size_t bytes) {
    size_t p = off;
    off = (off + bytes + 255) & ~(size_t)255;
    return p;
  };
  float*    bestIou    = (float*)(ws + take(BA * sizeof(float)));
  int*      bestIdx    = (int*)(ws + take(BA * sizeof(int)));
  float*    ceArr      = (float*)(ws + take(BA * sizeof(float)));
  unsigned* mkey       = (unsigned*)(ws + take(BA * sizeof(unsigned)));
  int*      bestAnchor = (int*)(ws + take((size_t)NB * NG * sizeof(int)));
  unsigned* posCount   = (unsigned*)(ws + take(NB * sizeof(unsigned)));
  unsigned* kthKey     = (unsigned*)(ws + take(NB * sizeof(unsigned)));
  unsigned* kPos       = (unsigned*)(ws + take(NB * sizeof(unsigned)));
  unsigned* totalPos   = (unsigned*)(ws + take(sizeof(unsigned)));
  float*    locSum     = (float*)(ws + take(sizeof(float)));
  float*    classSum   = (float*)(ws + take(sizeof(float)));
  (void)in_sizes; (void)n_in; (void)out_size; (void)ws_size;

  dim3 gA((NA + 255) / 256, NB);

  k_init<<<1, 64, 0, stream>>>(posCount, totalPos, locSum, classSum);
  k_match_anchor<<<gA, 256, 0, stream>>>(gts, counts, anchors, bestIou, bestIdx);
  k_best_anchor<<<dim3(NG, NB), 256, 0, stream>>>(gts, counts, anchors, bestAnchor);
  k_scatter<<<NB, 32, 0, stream>>>(counts, bestAnchor, bestIou, bestIdx);
  k_locpos<<<gA, 256, 0, stream>>>(gts, anchors, ploc, bestIou, bestIdx, posCount,
                                   totalPos, locSum);
  k_lse<<<dim3((NA + 7) / 8, NB), 256, 0, stream>>>(conf, gts, bestIou, bestIdx,
                                                    ceArr, mkey);
  k_select<<<NB, 256, 0, stream>>>(mkey, posCount, kthKey, kPos);
  k_classsum<<<gA, 256, 0, stream>>>(ceArr, mkey, kthKey, kPos, classSum);
  k_final<<<1, 32, 0, stream>>>(totalPos, locSum, classSum, out);
}